// SimGNN_76175539962368
// MI455X (gfx1250) — compile-verified
//
#include <hip/hip_runtime.h>
#include <hip/hip_bf16.h>

typedef float v2f __attribute__((ext_vector_type(2)));
typedef float v8f __attribute__((ext_vector_type(8)));

#define NNODES 256
#define NEDGES 2048
#define NB     128   // batch
#define NG     256   // total graphs (q + c)

// ---------------------------------------------------------------------------
// order-preserving float <-> uint encoding for global atomic min/max
// ---------------------------------------------------------------------------
__device__ __forceinline__ unsigned f2ord(float f) {
    unsigned u = __float_as_uint(f);
    return (u & 0x80000000u) ? ~u : (u | 0x80000000u);
}
__device__ __forceinline__ float ord2f(unsigned e) {
    unsigned u = (e & 0x80000000u) ? (e ^ 0x80000000u) : ~e;
    return __uint_as_float(u);
}

// ---------------------------------------------------------------------------
// init tiny buffers (min/max + histogram)
// ---------------------------------------------------------------------------
__global__ void init_kernel(unsigned* __restrict__ mm, float* __restrict__ hist) {
    int tid = threadIdx.x;
    if (tid == 0) { mm[0] = 0xFFFFFFFFu; mm[1] = 0u; }
    if (tid < 16) hist[tid] = 0.0f;
}

// ---------------------------------------------------------------------------
// per-graph degree -> dinv = rsqrt(deg + 1 self loop)
// ---------------------------------------------------------------------------
__global__ void compute_dinv(const int* __restrict__ eq, const int* __restrict__ ec,
                             float* __restrict__ dinv) {
    __shared__ unsigned cnt[NNODES];
    const int g = blockIdx.x, tid = threadIdx.x;
    cnt[tid] = 0u;
    __syncthreads();
    const int* e = (g < NB) ? (eq + (size_t)g * 2 * NEDGES)
                            : (ec + (size_t)(g - NB) * 2 * NEDGES);
    const int* dst = e + NEDGES;
    for (int i = tid; i < NEDGES; i += NNODES) atomicAdd(&cnt[dst[i]], 1u);
    __syncthreads();
    dinv[g * NNODES + tid] = rsqrtf(1.0f + (float)cnt[tid]);
}

// ---------------------------------------------------------------------------
// batched GEMM:  T[g] = Hin[g](256 x K) @ W(K x Nf)   via V_WMMA_F32_16X16X4_F32
// grid: (4 row-blocks of 64, NG graphs); block: 256 threads = 8 waves
// wave w: row-tile (w>>1), column half (w&1), NT = Nf/32 16x16 tiles
// ---------------------------------------------------------------------------
template <int K, int Nf>
__global__ void gcn_gemm(const float* __restrict__ Hq, const float* __restrict__ Hc,
                         const float* __restrict__ W, float* __restrict__ T) {
    extern __shared__ float smem[];                 // K*Nf staged weights
    const int g   = blockIdx.y;
    const int tid = threadIdx.x;
    const float* Hin = (g < NB) ? (Hq + (size_t)g * NNODES * K)
                                : (Hc + (size_t)(g - NB) * NNODES * K);
    for (int i = tid; i < K * Nf; i += 256) smem[i] = W[i];
    __syncthreads();

    const int wave = tid >> 5, lane = tid & 31;
    const int rt   = wave >> 1, half = wave & 1;
    const int rowBase = blockIdx.x * 64 + rt * 16;
    const int colBase = half * (Nf >> 1);
    constexpr int NT = Nf >> 5;                     // 4 / 2 / 1 tiles per wave
    const int m    = lane & 15;
    const int ksel = (lane >> 4) << 1;              // 0 or 2

    v8f acc[NT] = {};
    for (int kk = 0; kk < K; kk += 4) {
        const float* ap = Hin + (size_t)(rowBase + m) * K + kk + ksel;
        v2f a; a.x = ap[0]; a.y = ap[1];            // A[m][kk+ksel .. +1]
#pragma unroll
        for (int t = 0; t < NT; ++t) {
            const int n = colBase + t * 16 + m;
            v2f b;
            b.x = smem[(kk + ksel)     * Nf + n];   // B[k][n] = W[k][n]
            b.y = smem[(kk + ksel + 1) * Nf + n];
            acc[t] = __builtin_amdgcn_wmma_f32_16x16x4_f32(
                false, a, false, b, (short)0, acc[t], false, false);
        }
    }
    const int rsel = (lane >> 4) * 8;
    float* Tg = T + (size_t)g * NNODES * Nf;
#pragma unroll
    for (int t = 0; t < NT; ++t)
#pragma unroll
        for (int i = 0; i < 8; ++i) {
            const int row = rowBase + i + rsel;
            const int col = colBase + t * 16 + m;
            Tg[(size_t)row * Nf + col] = acc[t][i];
        }
}

// ---------------------------------------------------------------------------
// GCN combine: Hout = [relu](scatter_add(T[src]*coef -> dst) + T*dinv^2 + b)
// one block per graph; aggregation matrix lives in LDS (ds_add_f32 scatter)
// ---------------------------------------------------------------------------
template <int Nf, bool RELU>
__global__ void gcn_combine(const float* __restrict__ T, const float* __restrict__ dinv,
                            const int* __restrict__ eq, const int* __restrict__ ec,
                            const float* __restrict__ bias, float* __restrict__ Hout) {
    constexpr int CW = (Nf > 64) ? 64 : Nf;         // feature chunk width
    __shared__ float agg[NNODES * CW];
    __shared__ int   s_src[NEDGES];
    __shared__ int   s_dst[NEDGES];
    __shared__ float s_coef[NEDGES];

    const int g = blockIdx.x, tid = threadIdx.x;
    const int* e = (g < NB) ? (eq + (size_t)g * 2 * NEDGES)
                            : (ec + (size_t)(g - NB) * 2 * NEDGES);
    const int* src = e;
    const int* dst = e + NEDGES;
    const float* dv = dinv + g * NNODES;

    for (int i = tid; i < NEDGES; i += 256) {
        const int s = src[i], d = dst[i];
        s_src[i] = s; s_dst[i] = d;
        s_coef[i] = dv[s] * dv[d];
    }
    const float* Tg = T    + (size_t)g * NNODES * Nf;
    float*       Hg = Hout + (size_t)g * NNODES * Nf;

    for (int c0 = 0; c0 < Nf; c0 += CW) {
        __syncthreads();
        for (int i = tid; i < NNODES * CW; i += 256) agg[i] = 0.0f;
        __syncthreads();
        for (int i = tid; i < NEDGES * CW; i += 256) {
            const int eidx = i / CW, f = i % CW;
            const float v = Tg[(size_t)s_src[eidx] * Nf + c0 + f] * s_coef[eidx];
            atomicAdd(&agg[s_dst[eidx] * CW + f], v);
        }
        __syncthreads();
        for (int i = tid; i < NNODES * CW; i += 256) {
            const int n = i / CW, f = i % CW;
            const float d = dv[n];
            float h = agg[i] + Tg[(size_t)n * Nf + c0 + f] * d * d + bias[c0 + f];
            if (RELU) h = fmaxf(h, 0.0f);
            Hg[(size_t)n * Nf + c0 + f] = h;
        }
    }
}

// ---------------------------------------------------------------------------
// SimGNN attention pooling: E[g] = sum_n x[n] * sigmoid(x[n].ctx)
// ---------------------------------------------------------------------------
__global__ void att_pool(const float* __restrict__ H, const float* __restrict__ Watt,
                         float* __restrict__ E) {
    __shared__ float sumf[32], ctx[32], ef[32];
    const int g = blockIdx.x, tid = threadIdx.x;
    if (tid < 32) { sumf[tid] = 0.0f; ef[tid] = 0.0f; }
    __syncthreads();
    const float* Hg = H + (size_t)g * NNODES * 32;
    float x[32];
#pragma unroll
    for (int f = 0; f < 32; ++f) x[f] = Hg[tid * 32 + f];
#pragma unroll
    for (int f = 0; f < 32; ++f) atomicAdd(&sumf[f], x[f]);
    __syncthreads();
    if (tid < 32) {
        float c = 0.0f;
        for (int j = 0; j < 32; ++j) c += sumf[j] * Watt[tid * 32 + j];
        ctx[tid] = tanhf(c * (1.0f / (float)NNODES));
    }
    __syncthreads();
    float dot = 0.0f;
#pragma unroll
    for (int f = 0; f < 32; ++f) dot += x[f] * ctx[f];
    const float sc = 1.0f / (1.0f + expf(-dot));
#pragma unroll
    for (int f = 0; f < 32; ++f) atomicAdd(&ef[f], x[f] * sc);
    __syncthreads();
    if (tid < 32) E[g * 32 + tid] = ef[tid];
}

// ---------------------------------------------------------------------------
// NTN: relu(e1 . (W_t @ e2) + V_t . [e1;e2] + b_t)
// ---------------------------------------------------------------------------
__global__ void ntn_kernel(const float* __restrict__ E, const float* __restrict__ ntnW,
                           const float* __restrict__ ntnV, const float* __restrict__ ntnb,
                           float* __restrict__ S) {
    const int idx = blockIdx.x * blockDim.x + threadIdx.x;
    if (idx >= NB * 16) return;
    const int b = idx >> 4, t = idx & 15;
    const float* e1 = E + b * 32;
    const float* e2 = E + (NB + b) * 32;
    const float* Wt = ntnW + (size_t)t * 32 * 32;
    float acc = ntnb[t];
    for (int i = 0; i < 32; ++i) {
        float wi = 0.0f;
        for (int j = 0; j < 32; ++j) wi += Wt[i * 32 + j] * e2[j];
        acc += e1[i] * wi;
    }
    const float* Vt = ntnV + t * 64;
    for (int i = 0; i < 32; ++i) acc += Vt[i]      * e1[i];
    for (int j = 0; j < 32; ++j) acc += Vt[32 + j] * e2[j];
    S[idx] = fmaxf(acc, 0.0f);
}

// ---------------------------------------------------------------------------
// pairwise dots S[b] = Q[b](256x32) @ C[b]^T via WMMA + global min/max
// grid: (4 row-blocks, NB); block 256 = 8 waves; wave: row-tile (w>>1),
// column half (w&1) -> 8 tiles of 16 columns each
// ---------------------------------------------------------------------------
__global__ void pair_gemm(const float* __restrict__ Q, const float* __restrict__ C,
                          float* __restrict__ S, unsigned* __restrict__ mm) {
    __shared__ float redmin[256], redmax[256];
    const int b = blockIdx.y, tid = threadIdx.x;
    const int wave = tid >> 5, lane = tid & 31;
    const int rowBase = blockIdx.x * 64 + (wave >> 1) * 16;
    const int colBase = (wave & 1) * 128;
    const int m    = lane & 15;
    const int ksel = (lane >> 4) << 1;
    const float* Qb = Q + (size_t)b * NNODES * 32;
    const float* Cb = C + (size_t)b * NNODES * 32;

    v8f acc[8] = {};
    for (int kk = 0; kk < 32; kk += 4) {
        const float* ap = Qb + (size_t)(rowBase + m) * 32 + kk + ksel;
        v2f a; a.x = ap[0]; a.y = ap[1];
#pragma unroll
        for (int t = 0; t < 8; ++t) {
            const int n = colBase + t * 16 + m;
            const float* bp = Cb + (size_t)n * 32 + kk + ksel;  // B[k][n] = C[n][k]
            v2f bb; bb.x = bp[0]; bb.y = bp[1];
            acc[t] = __builtin_amdgcn_wmma_f32_16x16x4_f32(
                false, a, false, bb, (short)0, acc[t], false, false);
        }
    }
    float lmin = INFINITY, lmax = -INFINITY;
    const int rsel = (lane >> 4) * 8;
    float* Sb = S + (size_t)b * NNODES * NNODES;
#pragma unroll
    for (int t = 0; t < 8; ++t)
#pragma unroll
        for (int i = 0; i < 8; ++i) {
            const int row = rowBase + i + rsel;
            const int col = colBase + t * 16 + m;
            const float v = acc[t][i];
            Sb[(size_t)row * NNODES + col] = v;
            lmin = fminf(lmin, v);
            lmax = fmaxf(lmax, v);
        }
    redmin[tid] = lmin; redmax[tid] = lmax;
    __syncthreads();
    for (int s2 = 128; s2 > 0; s2 >>= 1) {
        if (tid < s2) {
            redmin[tid] = fminf(redmin[tid], redmin[tid + s2]);
            redmax[tid] = fmaxf(redmax[tid], redmax[tid + s2]);
        }
        __syncthreads();
    }
    if (tid == 0) {
        atomicMin(&mm[0], f2ord(redmin[0]));
        atomicMax(&mm[1], f2ord(redmax[0]));
    }
}

// ---------------------------------------------------------------------------
// 16-bin histogram (torch.histc semantics) over all pairwise dots
// ---------------------------------------------------------------------------
__global__ void hist_kernel(const float* __restrict__ S, const unsigned* __restrict__ mm,
                            float* __restrict__ hist, int total) {
    __shared__ unsigned lh[16];
    const int tid = threadIdx.x;
    if (tid < 16) lh[tid] = 0u;
    __syncthreads();
    const float lo = ord2f(mm[0]);
    const float hi = ord2f(mm[1]);
    const float scale = 16.0f / (hi - lo);
    const int base = blockIdx.x * blockDim.x * 16 + tid;
#pragma unroll
    for (int it = 0; it < 16; ++it) {
        const int i = base + it * blockDim.x;
        if (i < total) {
            int bn = (int)floorf((S[i] - lo) * scale);
            bn = min(15, max(0, bn));
            atomicAdd(&lh[bn], 1u);
        }
    }
    __syncthreads();
    if (tid < 16) atomicAdd(&hist[tid], (float)lh[tid]);
}

// ---------------------------------------------------------------------------
// final MLP head: [ntn_scores ; normalized hist] -> fc1 relu -> fc2 sigmoid
// ---------------------------------------------------------------------------
__global__ void head_kernel(const float* __restrict__ ntns, const float* __restrict__ hist,
                            const float* __restrict__ fc1W, const float* __restrict__ fc1b,
                            const float* __restrict__ fc2W, const float* __restrict__ fc2b,
                            float* __restrict__ out) {
    __shared__ float hn[16];
    const int tid = threadIdx.x;   // 128 threads
    if (tid == 0) {
        float tot = 0.0f;
        for (int i = 0; i < 16; ++i) tot += hist[i];
        const float inv = 1.0f / tot;
        for (int i = 0; i < 16; ++i) hn[i] = hist[i] * inv;
    }
    __syncthreads();
    float in[32];
#pragma unroll
    for (int t = 0; t < 16; ++t) in[t] = ntns[tid * 16 + t];
#pragma unroll
    for (int j = 0; j < 16; ++j) in[16 + j] = hn[j];
    float s = fc2b[0];
    for (int o = 0; o < 16; ++o) {
        float a = fc1b[o];
        for (int j = 0; j < 32; ++j) a += fc1W[o * 32 + j] * in[j];
        a = fmaxf(a, 0.0f);
        s += fc2W[o] * a;
    }
    out[tid] = 1.0f / (1.0f + expf(-s));
}

// ---------------------------------------------------------------------------
extern "C" void kernel_launch(void* const* d_in, const int* in_sizes, int n_in,
                              void* d_out, int out_size, void* d_ws, size_t ws_size,
                              hipStream_t stream) {
    const float* xq   = (const float*)d_in[0];
    const float* xc   = (const float*)d_in[1];
    const int*   eq   = (const int*)  d_in[2];
    const int*   ec   = (const int*)  d_in[3];
    const float* W1   = (const float*)d_in[4];
    const float* b1   = (const float*)d_in[5];
    const float* W2   = (const float*)d_in[6];
    const float* b2   = (const float*)d_in[7];
    const float* W3   = (const float*)d_in[8];
    const float* b3   = (const float*)d_in[9];
    const float* Watt = (const float*)d_in[10];
    const float* ntnW = (const float*)d_in[11];
    const float* ntnV = (const float*)d_in[12];
    const float* ntnb = (const float*)d_in[13];
    const float* fc1W = (const float*)d_in[14];
    const float* fc1b = (const float*)d_in[15];
    const float* fc2W = (const float*)d_in[16];
    const float* fc2b = (const float*)d_in[17];
    float* out = (float*)d_out;

    char* ws = (char*)d_ws;
    size_t off = 0;
    auto alloc = [&](size_t bytes) -> void* {
        void* p = ws + off;
        off = (off + bytes + 255) & ~(size_t)255;
        return p;
    };
    float*    dinv = (float*)   alloc((size_t)NG * NNODES * 4);        // 256 KB
    float*    t    = (float*)   alloc((size_t)NG * NNODES * 128 * 4);  // 32 MB
    float*    h1   = (float*)   alloc((size_t)NG * NNODES * 128 * 4);  // 32 MB
    float*    h2   = (float*)   alloc((size_t)NG * NNODES * 64  * 4);  // 16 MB
    float*    h3   = (float*)   alloc((size_t)NG * NNODES * 32  * 4);  // 8 MB
    float*    E    = (float*)   alloc((size_t)NG * 32 * 4);
    float*    ntns = (float*)   alloc((size_t)NB * 16 * 4);
    unsigned* mm   = (unsigned*)alloc(256);
    float*    hist = (float*)   alloc(256);
    float*    S    = t;   // pairwise-dot buffer aliases the (dead) GEMM temp

    init_kernel<<<1, 64, 0, stream>>>(mm, hist);
    compute_dinv<<<NG, NNODES, 0, stream>>>(eq, ec, dinv);

    // layer 1: 128 -> 128, relu
    gcn_gemm<128, 128><<<dim3(4, NG), 256, 128 * 128 * 4, stream>>>(xq, xc, W1, t);
    gcn_combine<128, true><<<NG, 256, 0, stream>>>(t, dinv, eq, ec, b1, h1);
    // layer 2: 128 -> 64, relu
    gcn_gemm<128, 64><<<dim3(4, NG), 256, 128 * 64 * 4, stream>>>(
        h1, h1 + (size_t)NB * NNODES * 128, W2, t);
    gcn_combine<64, true><<<NG, 256, 0, stream>>>(t, dinv, eq, ec, b2, h2);
    // layer 3: 64 -> 32, no relu
    gcn_gemm<64, 32><<<dim3(4, NG), 256, 64 * 32 * 4, stream>>>(
        h2, h2 + (size_t)NB * NNODES * 64, W3, t);
    gcn_combine<32, false><<<NG, 256, 0, stream>>>(t, dinv, eq, ec, b3, h3);

    att_pool<<<NG, 256, 0, stream>>>(h3, Watt, E);
    ntn_kernel<<<8, 256, 0, stream>>>(E, ntnW, ntnV, ntnb, ntns);

    pair_gemm<<<dim3(4, NB), 256, 0, stream>>>(
        h3, h3 + (size_t)NB * NNODES * 32, S, mm);
    const int total = NB * NNODES * NNODES;   // 8388608
    hist_kernel<<<total / (256 * 16), 256, 0, stream>>>(S, mm, hist, total);

    head_kernel<<<1, NB, 0, stream>>>(ntns, hist, fc1W, fc1b, fc2W, fc2b, out);
}